// LightGCN_14379550507255
// MI455X (gfx1250) — compile-verified
//
#include <hip/hip_runtime.h>

#define NUM_USERS_ 100000
#define NUM_ITEMS_ 50000
#define N_NODES_   (NUM_USERS_ + NUM_ITEMS_)
#define DIM_       64
#define NNZ_       2400000
#define BATCH_     16384
#define ND_        (N_NODES_ * DIM_)     /* 9,600,000 floats */
#define UND_       (NUM_USERS_ * DIM_)   /* 6,400,000 floats */

typedef __attribute__((ext_vector_type(2))) float v2f;
typedef __attribute__((ext_vector_type(8))) float v8f;

// ---- emb0 = concat(user_emb, item_emb); acc = emb0 -------------------------
__global__ void lgcn_init_concat(const float4* __restrict__ u,
                                 const float4* __restrict__ it,
                                 float4* __restrict__ emb,
                                 float4* __restrict__ acc) {
  int i = blockIdx.x * blockDim.x + threadIdx.x;
  if (i >= ND_ / 4) return;
  float4 v = (i < UND_ / 4) ? u[i] : it[i - UND_ / 4];
  emb[i] = v;
  acc[i] = v;
}

// ---- zero a float4 buffer ---------------------------------------------------
__global__ void lgcn_zero(float4* __restrict__ p, int n4) {
  int i = blockIdx.x * blockDim.x + threadIdx.x;
  if (i < n4) p[i] = make_float4(0.f, 0.f, 0.f, 0.f);
}

// ---- y[rows[e]] += vals[e] * x[cols[e]]  (16 threads per edge, float4) -----
__global__ void lgcn_spmm_scatter(const float* __restrict__ vals,
                                  const int* __restrict__ rows,
                                  const int* __restrict__ cols,
                                  const float* __restrict__ x,
                                  float* __restrict__ y) {
  int t = blockIdx.x * blockDim.x + threadIdx.x;   // < NNZ_*16 = 38.4M
  int e = t >> 4;
  if (e >= NNZ_) return;
  int q = (t & 15) << 2;                           // dim offset 0..60
  float v = vals[e];
  int c = cols[e];
  int r = rows[e];
  const float4 src = *(const float4*)(x + (long long)c * DIM_ + q);
  float* dst = y + (long long)r * DIM_ + q;
  atomicAdd(dst + 0, v * src.x);
  atomicAdd(dst + 1, v * src.y);
  atomicAdd(dst + 2, v * src.z);
  atomicAdd(dst + 3, v * src.w);
}

// ---- acc += next ------------------------------------------------------------
__global__ void lgcn_acc_add(float4* __restrict__ acc,
                             const float4* __restrict__ nx, int n4) {
  int i = blockIdx.x * blockDim.x + threadIdx.x;
  if (i >= n4) return;
  float4 a = acc[i];
  float4 b = nx[i];
  a.x += b.x; a.y += b.y; a.z += b.z; a.w += b.w;
  acc[i] = a;
}

// ---- gamma[b] = dot(acc[users[b]], acc[NUM_USERS+items[b]]) / 16 -----------
// One wave handles 16 batch rows: S(16x16) = U(16x64) @ I^T(64x16) via 16
// chained V_WMMA_F32_16X16X4_F32 ops; gamma = diag(S) * (1/16).
// Layouts per CDNA5 ISA 7.12.2 (f32, wave32):
//   A 16x4 : lane m(=lane&15) row M=m; VGPR0/1 = K{0,1} (lanes 0-15) or K{2,3}
//   B 4x16 : lane m col N=m;   VGPR0/1 = K{0,1} (lanes 0-15) or K{2,3}
//   D 16x16: VGPR r, lanes 0-15 -> M=r, lanes 16-31 -> M=r+8
__global__ void lgcn_score_wmma(const float* __restrict__ acc,
                                const int* __restrict__ users,
                                const int* __restrict__ items,
                                float* __restrict__ out) {
  int wave = (blockIdx.x * blockDim.x + threadIdx.x) >> 5;   // 0..1023
  int lane = threadIdx.x & 31;
  int base = wave * 16;                                      // 16 rows/wave
  int m    = lane & 15;
  int half = lane >> 4;

  int uIdx = users[base + m];
  int iIdx = items[base + m];
  const float* rowU = acc + (long long)uIdx * DIM_;
  const float* rowI = acc + (long long)(NUM_USERS_ + iIdx) * DIM_;

  v8f c = {0.f, 0.f, 0.f, 0.f, 0.f, 0.f, 0.f, 0.f};
#pragma unroll
  for (int k0 = 0; k0 < DIM_; k0 += 4) {
    int off = k0 + 2 * half;
    v2f a; a.x = rowU[off]; a.y = rowU[off + 1];
    v2f b; b.x = rowI[off]; b.y = rowI[off + 1];
    // 8 args: (neg_a, A, neg_b, B, c_mod, C, reuse_a, reuse_b)
    c = __builtin_amdgcn_wmma_f32_16x16x4_f32(false, a, false, b,
                                              (short)0, c, false, false);
  }

  // Diagonal extraction: D[j][j] lives at (lane j, c[j]) for j<8 and at
  // (lane j+16, c[j-8]) for j>=8.
  float diag = 0.f;
#pragma unroll
  for (int j = 0; j < 8; ++j) {
    if (lane == j)      diag = c[j];
    if (lane == j + 24) diag = c[j];
  }
  const float scale = 1.0f / 16.0f;   // (1/(N_LAYERS+1))^2 = (1/4)^2
  if (lane < 8)       out[base + lane]      = diag * scale;
  else if (lane >= 24) out[base + lane - 16] = diag * scale;
}

extern "C" void kernel_launch(void* const* d_in, const int* in_sizes, int n_in,
                              void* d_out, int out_size, void* d_ws, size_t ws_size,
                              hipStream_t stream) {
  (void)in_sizes; (void)n_in; (void)out_size; (void)ws_size;
  const float* user_emb = (const float*)d_in[0];
  const float* item_emb = (const float*)d_in[1];
  const float* adj_vals = (const float*)d_in[2];
  const int*   adj_rows = (const int*)d_in[3];
  const int*   adj_cols = (const int*)d_in[4];
  const int*   users    = (const int*)d_in[5];
  const int*   items    = (const int*)d_in[6];
  float* out = (float*)d_out;

  // Workspace: 3 buffers of ND_ floats (emb A/B ping-pong + running acc).
  float* embA = (float*)d_ws;
  float* embB = embA + (size_t)ND_;
  float* acc  = embA + 2 * (size_t)ND_;

  const int B = 256;
  const int n4 = ND_ / 4;
  const int gElem = (n4 + B - 1) / B;

  lgcn_init_concat<<<gElem, B, 0, stream>>>(
      (const float4*)user_emb, (const float4*)item_emb,
      (float4*)embA, (float4*)acc);

  float* cur = embA;
  float* nxt = embB;
  for (int layer = 0; layer < 3; ++layer) {
    lgcn_zero<<<gElem, B, 0, stream>>>((float4*)nxt, n4);
    int scatterThreads = NNZ_ * 16;                       // 38.4M
    lgcn_spmm_scatter<<<(scatterThreads + B - 1) / B, B, 0, stream>>>(
        adj_vals, adj_rows, adj_cols, cur, nxt);
    lgcn_acc_add<<<gElem, B, 0, stream>>>((float4*)acc, (const float4*)nxt, n4);
    float* t = cur; cur = nxt; nxt = t;                   // deterministic swap
  }

  // 16384/16 = 1024 waves -> 32768 threads -> 128 blocks of 256
  lgcn_score_wmma<<<(BATCH_ / 16) * 32 / B, B, 0, stream>>>(acc, users, items, out);
}